// Attn_head_46420006535794
// MI455X (gfx1250) — compile-verified
//
#include <hip/hip_runtime.h>
#include <stdint.h>

// ---------------------------------------------------------------------------
// GAT attention head for MI455X (gfx1250, wave32).
//  - dual GEMM on the WMMA f32 pipe (V_WMMA_F32_16X16X4_F32): exact fp32,
//    and the GEMM is HBM-trivial (~1.6 GFLOP, 26 MB) so low precision buys
//    nothing here.
//  - edge phase is L2-resident random gather + hardware fp32 atomics
//    (global_atomic_add_f32), which is what dominates runtime.
// ---------------------------------------------------------------------------

typedef __attribute__((ext_vector_type(2))) float v2f;
typedef __attribute__((ext_vector_type(8))) float v8f;

#define IN_CH       128
#define OUT_SZ      64
#define GEMM_BLOCKS 400   // 8 waves/block -> 3200 waves; 8 col tiles -> 400 row-tile streams

__device__ __forceinline__ unsigned enc_f32(float x) {
  unsigned u = __float_as_uint(x);
  return (u & 0x80000000u) ? ~u : (u | 0x80000000u);   // monotonic float->uint
}
__device__ __forceinline__ float dec_f32(unsigned m) {
  unsigned u = (m & 0x80000000u) ? (m & 0x7fffffffu) : ~m;
  return __uint_as_float(u);
}
__device__ __forceinline__ void atomic_add_f32(float* p, float v) {
  __hip_atomic_fetch_add(p, v, __ATOMIC_RELAXED, __HIP_MEMORY_SCOPE_AGENT);
}

// ---------------------------------------------------------------------------
// K0: reset reduction scalars (graph-capture safe; no memsets needed).
__global__ void init_red(unsigned* gmax_bits, float* gsum) {
  if (threadIdx.x == 0) { *gmax_bits = 0u; *gsum = 0.0f; }
}

// ---------------------------------------------------------------------------
// K1: dual GEMM.  Wave w: col tile ct = w&7 (ct<4 -> W_seq cols, ct>=4 -> W_res
// cols), row tiles rt = w>>3, w>>3 + 400, ...
// f32 WMMA 16x16x4 layouts (ISA 7.12.2):
//   A 16x4 : lane l holds row M=l&15, K pair = 2*(l>>4) .. +1  (v2f)
//   B 4x16 : lane l holds col N=l&15, K pair = 2*(l>>4) .. +1  (v2f)
//   C/D    : lane l col N=l&15; VGPR r holds row M = r + 8*(l>>4)
__global__ void gemm_wmma(const float* __restrict__ seq,
                          const float* __restrict__ Wseq,
                          const float* __restrict__ Wres,
                          const float* __restrict__ bias,
                          const float* __restrict__ bres,
                          float* __restrict__ seq_fts,
                          float* __restrict__ out,
                          int nrowtiles, int nwaves) {
  const int wave = blockIdx.x * (blockDim.x >> 5) + (threadIdx.x >> 5);
  const int lane = threadIdx.x & 31;
  const int ct   = wave & 7;
  const int rstride = nwaves >> 3;

  const float* W = (ct < 4) ? Wseq : Wres;
  const int colbase = (ct & 3) * 16;
  const int nloc  = lane & 15;          // B column / D column / A row (within tile)
  const int kh    = (lane >> 4) * 2;    // K sub-pair selected by lane half
  const int rhalf = (lane >> 4) * 8;    // D row offset selected by lane half

  // Preload the wave's B panel: 32 k-steps of K=4 -> 32 x v2f (stays in VGPRs).
  v2f breg[32];
  const float* wrow = W + (size_t)(colbase + nloc) * IN_CH + kh;
  #pragma unroll
  for (int s = 0; s < 32; ++s) breg[s] = *(const v2f*)(wrow + 4 * s);

  for (int rt = wave >> 3; rt < nrowtiles; rt += rstride) {
    const int rowbase = rt * 16;
    const float* arow = seq + (size_t)(rowbase + nloc) * IN_CH + kh;
    // Speculative prefetch of the next row tile (global_prefetch_b8; dropped
    // silently if it runs past the buffer).
    __builtin_prefetch(arow + (size_t)rstride * 16 * IN_CH, 0, 1);

    v8f acc = {};
    #pragma unroll
    for (int s = 0; s < 32; ++s) {
      v2f a = *(const v2f*)(arow + 4 * s);
      acc = __builtin_amdgcn_wmma_f32_16x16x4_f32(
          /*neg_a=*/false, a, /*neg_b=*/false, breg[s],
          /*c_mod=*/(short)0, acc, /*reuse_a=*/false, /*reuse_b=*/false);
    }

    if (ct < 4) {
      #pragma unroll
      for (int r = 0; r < 8; ++r)
        seq_fts[(size_t)(rowbase + rhalf + r) * OUT_SZ + colbase + nloc] = acc[r];
    } else {
      const float badd = bres[colbase + nloc] + bias[colbase + nloc];
      #pragma unroll
      for (int r = 0; r < 8; ++r)
        out[(size_t)(rowbase + rhalf + r) * OUT_SZ + colbase + nloc] = acc[r] + badd;
    }
  }
}

// ---------------------------------------------------------------------------
// K2: f1[n] = seq_fts[n,:]·w_f1 + b_f1 ; f2 likewise.
__global__ void node_scores(const float* __restrict__ seq_fts,
                            const float* __restrict__ wf1, const float* __restrict__ bf1,
                            const float* __restrict__ wf2, const float* __restrict__ bf2,
                            float* __restrict__ f1, float* __restrict__ f2, int N) {
  int n = blockIdx.x * blockDim.x + threadIdx.x;
  if (n >= N) return;
  const float* row = seq_fts + (size_t)n * OUT_SZ;
  float s1 = 0.f, s2 = 0.f;
  #pragma unroll
  for (int c = 0; c < OUT_SZ; c += 4) {
    float4 v = *(const float4*)(row + c);
    float4 a = *(const float4*)(wf1 + c);
    float4 b = *(const float4*)(wf2 + c);
    s1 += v.x * a.x + v.y * a.y + v.z * a.z + v.w * a.w;
    s2 += v.x * b.x + v.y * b.y + v.z * b.z + v.w * b.w;
  }
  f1[n] = s1 + bf1[0];
  f2[n] = s2 + bf2[0];
}

// ---------------------------------------------------------------------------
// K3: e = leaky_relu(f1[src]+f2[dst]); track global max (encoded atomicMax).
__global__ void edge_logits(const int* __restrict__ ei,
                            const float* __restrict__ f1, const float* __restrict__ f2,
                            float* __restrict__ evals, unsigned* gmax_bits, int E) {
  __shared__ float smax[256];
  int e = blockIdx.x * blockDim.x + threadIdx.x;
  float v = -3.402823e38f;
  if (e < E) {
    int src = ei[e];
    int dst = ei[(size_t)E + e];
    float x = f1[src] + f2[dst];
    v = x > 0.f ? x : 0.01f * x;
    evals[e] = v;
  }
  smax[threadIdx.x] = v;
  __syncthreads();
  for (int off = 128; off > 0; off >>= 1) {
    if (threadIdx.x < (unsigned)off)
      smax[threadIdx.x] = fmaxf(smax[threadIdx.x], smax[threadIdx.x + off]);
    __syncthreads();
  }
  if (threadIdx.x == 0) atomicMax(gmax_bits, enc_f32(smax[0]));
}

// ---------------------------------------------------------------------------
// K4: evals[e] = exp(e - max); accumulate global sum.
__global__ void edge_exp_sum(float* __restrict__ evals, const unsigned* __restrict__ gmax_bits,
                             float* __restrict__ gsum, int E) {
  __shared__ float ssum[256];
  const float m = dec_f32(*gmax_bits);
  int e = blockIdx.x * blockDim.x + threadIdx.x;
  float ex = 0.f;
  if (e < E) {
    ex = __expf(evals[e] - m);
    evals[e] = ex;
  }
  ssum[threadIdx.x] = ex;
  __syncthreads();
  for (int off = 128; off > 0; off >>= 1) {
    if (threadIdx.x < (unsigned)off)
      ssum[threadIdx.x] += ssum[threadIdx.x + off];
    __syncthreads();
  }
  if (threadIdx.x == 0) atomic_add_f32(gsum, ssum[0]);
}

// ---------------------------------------------------------------------------
// K5: scatter: out[src,:] += (evals[e]/sum) * seq_fts[dst,:]
// 16 threads per edge, 4 channels each: float4 gather + 4 HW fp32 atomics.
__global__ void edge_scatter(const int* __restrict__ ei,
                             const float* __restrict__ evals,
                             const float* __restrict__ gsum,
                             const float* __restrict__ seq_fts,
                             float* __restrict__ out, int E) {
  long long tid = (long long)blockIdx.x * blockDim.x + threadIdx.x;
  int e = (int)(tid >> 4);
  if (e >= E) return;
  int c = (int)(tid & 15) << 2;
  int src = ei[e];
  int dst = ei[(size_t)E + e];
  float coef = evals[e] / gsum[0];
  float4 v = *(const float4*)(seq_fts + (size_t)dst * OUT_SZ + c);
  float* o = out + (size_t)src * OUT_SZ + c;
  atomic_add_f32(o + 0, coef * v.x);
  atomic_add_f32(o + 1, coef * v.y);
  atomic_add_f32(o + 2, coef * v.z);
  atomic_add_f32(o + 3, coef * v.w);
}

// ---------------------------------------------------------------------------
// K6: ELU in place (alpha = 1).
__global__ void elu_inplace(float* __restrict__ out, int n) {
  int i = blockIdx.x * blockDim.x + threadIdx.x;
  if (i < n) {
    float x = out[i];
    out[i] = x > 0.f ? x : (__expf(x) - 1.0f);
  }
}

// ---------------------------------------------------------------------------
extern "C" void kernel_launch(void* const* d_in, const int* in_sizes, int n_in,
                              void* d_out, int out_size, void* d_ws, size_t ws_size,
                              hipStream_t stream) {
  const float* seq  = (const float*)d_in[0];
  const int*   ei   = (const int*)  d_in[1];   // edge_index (2,E) ints
  const float* Wseq = (const float*)d_in[2];
  const float* wf1  = (const float*)d_in[3];
  const float* bf1  = (const float*)d_in[4];
  const float* wf2  = (const float*)d_in[5];
  const float* bf2  = (const float*)d_in[6];
  const float* bias = (const float*)d_in[7];
  const float* Wres = (const float*)d_in[8];
  const float* bres = (const float*)d_in[9];
  float* out = (float*)d_out;

  const int N = in_sizes[0] / IN_CH;
  const int E = in_sizes[1] / 2;

  // Workspace layout (floats): seq_fts[N*64] | f1[N] | f2[N] | evals[E] | gmax | gsum
  float* seq_fts = (float*)d_ws;
  float* f1      = seq_fts + (size_t)N * OUT_SZ;
  float* f2      = f1 + N;
  float* evals   = f2 + N;
  unsigned* gmax_bits = (unsigned*)(evals + E);
  float*    gsum      = (float*)(gmax_bits + 1);

  init_red<<<1, 64, 0, stream>>>(gmax_bits, gsum);

  const int nrowtiles = N / 16;              // N = 50000 -> 3125
  const int nwaves = GEMM_BLOCKS * 8;
  gemm_wmma<<<GEMM_BLOCKS, 256, 0, stream>>>(seq, Wseq, Wres, bias, bres,
                                             seq_fts, out, nrowtiles, nwaves);

  node_scores<<<(N + 255) / 256, 256, 0, stream>>>(seq_fts, wf1, bf1, wf2, bf2, f1, f2, N);

  edge_logits<<<(E + 255) / 256, 256, 0, stream>>>(ei, f1, f2, evals, gmax_bits, E);
  edge_exp_sum<<<(E + 255) / 256, 256, 0, stream>>>(evals, gmax_bits, gsum, E);

  long long scatter_threads = (long long)E * 16;
  edge_scatter<<<(unsigned)((scatter_threads + 255) / 256), 256, 0, stream>>>(
      ei, evals, gsum, seq_fts, out, E);

  elu_inplace<<<(N * OUT_SZ + 255) / 256, 256, 0, stream>>>(out, N * OUT_SZ);
}